// PTBLM_30434138259669
// MI455X (gfx1250) — compile-verified
//
#include <hip/hip_runtime.h>
#include <math.h>

// Problem constants (match reference)
#define V_  32000
#define E_  1024
#define H_  1024
#define B_  16
#define T_  128
#define FH_ 4096   // 4*H

typedef float v2f __attribute__((ext_vector_type(2)));
typedef float v4f __attribute__((ext_vector_type(4)));
typedef float v8f __attribute__((ext_vector_type(8)));

// D = A(16x4) * B(4x16) + C, fp32 WMMA (wave32). 8-arg form:
// (neg_a, A, neg_b, B, c_mod, C, reuse_a, reuse_b)
#define WMMA_F32(a, b, c) \
  __builtin_amdgcn_wmma_f32_16x16x4_f32(false, (a), false, (b), (short)0, (c), false, false)

__device__ __forceinline__ float sigm(float x) { return 1.0f / (1.0f + expf(-x)); }

// ---------------------------------------------------------------------------
// Embedding gather: x[(t*B + b), e] = emb[idx[b*T + t], e]   (T,B,E) layout
// ---------------------------------------------------------------------------
__global__ __launch_bounds__(256)
void embed_kernel(const int* __restrict__ idx, const float* __restrict__ emb,
                  float* __restrict__ x) {
  int i  = blockIdx.x * blockDim.x + threadIdx.x;   // over T*B*(E/4)
  int e4 = i % (E_ / 4);
  int tb = i / (E_ / 4);
  int t = tb / B_, b = tb % B_;
  const v4f* src = (const v4f*)(emb + (size_t)idx[b * T_ + t] * E_);
  ((v4f*)(x + (size_t)tb * E_))[e4] = src[e4];
}

// ---------------------------------------------------------------------------
// C = A(MxK,row) @ Bm(KxN,row) [+ D] [+ bias]; one wave per 16x64 strip.
// block = 32 threads; grid.x = N/64, grid.y = M/16.
// ---------------------------------------------------------------------------
__global__ __launch_bounds__(32)
void gemm_rr_wmma(const float* __restrict__ A, const float* __restrict__ Bm,
                  const float* __restrict__ D, const float* __restrict__ bias,
                  float* __restrict__ C, int K, int lda, int ldb, int ldc) {
  const int lane = threadIdx.x;
  const int lid  = lane & 15;
  const int hf   = lane >> 4;
  const int m0   = blockIdx.y * 16;
  const int n0   = blockIdx.x * 64;

  v8f acc[4] = {};
  const float* Arow = A + (size_t)(m0 + lid) * lda + 2 * hf;

  for (int k = 0; k < K; k += 4) {
    v2f a = *(const v2f*)(Arow + k);
    const float* Bk0 = Bm + (size_t)(k + 2 * hf) * ldb + n0 + lid;
    const float* Bk1 = Bk0 + ldb;
#pragma unroll
    for (int j = 0; j < 4; ++j) {
      v2f b;
      b.x = Bk0[16 * j];
      b.y = Bk1[16 * j];
      acc[j] = WMMA_F32(a, b, acc[j]);
    }
  }

#pragma unroll
  for (int j = 0; j < 4; ++j) {
    int col = n0 + 16 * j + lid;
    float bb = bias ? bias[col] : 0.0f;
#pragma unroll
    for (int v = 0; v < 8; ++v) {
      int row = m0 + v + 8 * hf;
      size_t off = (size_t)row * ldc + col;
      float val = acc[j][v] + bb;
      if (D) val += D[off];
      C[off] = val;
    }
  }
}

// ---------------------------------------------------------------------------
// Decoder: out[b, t, v] = sum_k h2[(t*B+b), k] * emb[v, k] + tie_b[v]
// emb acts as B-transposed (contiguous b64 per-lane loads).
// grid.x = V/64, grid.y = (T*B)/16.
// ---------------------------------------------------------------------------
__global__ __launch_bounds__(32)
void decoder_wmma(const float* __restrict__ h2, const float* __restrict__ emb,
                  const float* __restrict__ tie_b, float* __restrict__ out) {
  const int lane = threadIdx.x;
  const int lid  = lane & 15;
  const int hf   = lane >> 4;
  const int m0   = blockIdx.y * 16;   // over T*B
  const int n0   = blockIdx.x * 64;   // over V

  v8f acc[4] = {};
  const float* Arow = h2 + (size_t)(m0 + lid) * H_ + 2 * hf;
  const float* Brow = emb + (size_t)(n0 + lid) * H_ + 2 * hf;

  for (int k = 0; k < H_; k += 4) {
    v2f a = *(const v2f*)(Arow + k);
#pragma unroll
    for (int j = 0; j < 4; ++j) {
      v2f b = *(const v2f*)(Brow + (size_t)(16 * j) * H_ + k);
      acc[j] = WMMA_F32(a, b, acc[j]);
    }
  }

#pragma unroll
  for (int j = 0; j < 4; ++j) {
    int col = n0 + 16 * j + lid;
    float bb = tie_b[col];
#pragma unroll
    for (int v = 0; v < 8; ++v) {
      int row = m0 + v + 8 * hf;      // row = t*16 + b  (B==16)
      int t = row >> 4, b = row & 15;
      out[(size_t)b * ((size_t)T_ * V_) + (size_t)t * V_ + col] = acc[j][v] + bb;
    }
  }
}

// ---------------------------------------------------------------------------
// LSTM elementwise step: gates (B,4H) order [f, i, cn, o]; updates c,h in
// place and records h into the sequence slice for timestep t.
// ---------------------------------------------------------------------------
__global__ __launch_bounds__(256)
void lstm_elem(const float* __restrict__ gates, float* __restrict__ c,
               float* __restrict__ h, float* __restrict__ hseq_t) {
  int i = blockIdx.x * blockDim.x + threadIdx.x;  // B*H
  int b = i >> 10;          // H = 1024
  int j = i & (H_ - 1);
  const float* g = gates + (size_t)b * FH_;
  float f  = sigm(g[j]);
  float ii = sigm(g[H_ + j]);
  float cn = tanhf(g[2 * H_ + j]);
  float o  = sigm(g[3 * H_ + j]);
  float cv = c[i] * f + ii * cn;
  float hv = o * tanhf(cv);
  c[i] = cv;
  h[i] = hv;
  hseq_t[i] = hv;
}

__global__ __launch_bounds__(256)
void copy2_kernel(const float* __restrict__ a, const float* __restrict__ b,
                  float* __restrict__ da, float* __restrict__ db, int n) {
  int i = blockIdx.x * blockDim.x + threadIdx.x;
  if (i < n) { da[i] = a[i]; db[i] = b[i]; }
}

// ---------------------------------------------------------------------------
extern "C" void kernel_launch(void* const* d_in, const int* in_sizes, int n_in,
                              void* d_out, int out_size, void* d_ws, size_t ws_size,
                              hipStream_t stream) {
  const int*   idx   = (const int*)d_in[0];
  const float* ic    = (const float*)d_in[1];
  const float* ih    = (const float*)d_in[2];
  const float* emb   = (const float*)d_in[3];
  const float* tie_b = (const float*)d_in[4];
  const float* U0  = (const float*)d_in[5];
  const float* bU0 = (const float*)d_in[6];
  const float* W0  = (const float*)d_in[7];
  const float* bW0 = (const float*)d_in[8];
  const float* U1  = (const float*)d_in[9];
  const float* bU1 = (const float*)d_in[10];
  const float* W1  = (const float*)d_in[11];
  const float* bW1 = (const float*)d_in[12];
  float* out = (float*)d_out;
  float* ws  = (float*)d_ws;

  // Workspace layout (floats); x buffer is recycled for h2, xu for both layers.
  float* x     = ws;                                 // T*B*E  = 2M  (later h2 seq)
  float* xu    = x  + (size_t)T_ * B_ * E_;          // T*B*4H = 8M  (xU0 then xU1)
  float* h1    = xu + (size_t)T_ * B_ * FH_;         // T*B*H  = 2M
  float* gates = h1 + (size_t)T_ * B_ * H_;          // B*4H
  float* cs    = gates + (size_t)B_ * FH_;           // B*H
  float* hs    = cs + (size_t)B_ * H_;               // B*H

  const int BH = B_ * H_;
  const dim3 blk32(32), blk256(256);
  const dim3 grid_xu(FH_ / 64, (T_ * B_) / 16);      // (64,128)
  const dim3 grid_rec(FH_ / 64, 1);                  // (64,1)
  const dim3 grid_dec(V_ / 64, (T_ * B_) / 16);      // (500,128)

  // 1) embedding gather -> x (T,B,E)
  embed_kernel<<<(T_ * B_ * E_ / 4) / 256, blk256, 0, stream>>>(idx, emb, x);

  // 2) layer 0 input projection: xu = x @ U0 + bU0
  gemm_rr_wmma<<<grid_xu, blk32, 0, stream>>>(x, U0, nullptr, bU0, xu,
                                              E_, E_, FH_, FH_);

  // 3) layer 0 recurrence (initial states)
  copy2_kernel<<<(BH + 255) / 256, blk256, 0, stream>>>(ic, ih, cs, hs, BH);
  for (int t = 0; t < T_; ++t) {
    gemm_rr_wmma<<<grid_rec, blk32, 0, stream>>>(
        hs, W0, xu + (size_t)t * B_ * FH_, bW0, gates, H_, H_, FH_, FH_);
    lstm_elem<<<BH / 256, blk256, 0, stream>>>(gates, cs, hs,
                                               h1 + (size_t)t * B_ * H_);
  }

  // 4) layer 1 input projection: xu = h1 @ U1 + bU1
  gemm_rr_wmma<<<grid_xu, blk32, 0, stream>>>(h1, U1, nullptr, bU1, xu,
                                              H_, H_, FH_, FH_);

  // 5) layer 1 recurrence (same initial states, per reference); h2 -> x buffer
  copy2_kernel<<<(BH + 255) / 256, blk256, 0, stream>>>(ic, ih, cs, hs, BH);
  for (int t = 0; t < T_; ++t) {
    gemm_rr_wmma<<<grid_rec, blk32, 0, stream>>>(
        hs, W1, xu + (size_t)t * B_ * FH_, bW1, gates, H_, H_, FH_, FH_);
    lstm_elem<<<BH / 256, blk256, 0, stream>>>(gates, cs, hs,
                                               x + (size_t)t * B_ * H_);
  }

  // 6) tied decoder: logits (B,T,V) = h2 @ emb^T + tie_b
  decoder_wmma<<<grid_dec, blk32, 0, stream>>>(x, emb, tie_b, out);

  // 7) final (c, h) appended after logits
  size_t off = (size_t)B_ * T_ * V_;
  copy2_kernel<<<(BH + 255) / 256, blk256, 0, stream>>>(cs, hs, out + off,
                                                        out + off + BH, BH);
}